// LassomodPix_v2_62929860821463
// MI455X (gfx1250) — compile-verified
//
#include <hip/hip_runtime.h>

typedef __attribute__((ext_vector_type(2))) float v2f;
typedef __attribute__((ext_vector_type(8))) float v8f;

#define NX_G 82
#define NY_G 67
#define P_G  5494   // NX*NY pixels
#define F_G  64     // features (K)
#define B_G  8192   // batch (M)

// C[b, p] = sum_f A[b, f] * W[p, f];  out[b, x, y] = C[b, y*82 + x] + bias[p]
//
// Per wave32: 16 (M=batch) x 64 (N=pixel) output strip.
//   A fragments (fp32 16x4 WMMA layout): lane L holds row M = L%16,
//     K-pair = 2*(L/16) + {0,1}  -> contiguous 8B load from row-major A.
//   B fragments (fp32 4x16 WMMA layout): lane L holds col N = L%16,
//     K-pair = 2*(L/16) + {0,1}  -> contiguous 8B load from row-major W.
// K = 64 -> 16 chained v_wmma_f32_16x16x4_f32 per 16x16 tile, 64 per wave.
__global__ __launch_bounds__(256) void lasso_pix_gemm(
    const float* __restrict__ A,     // [B_G, F_G]
    const float* __restrict__ W,     // [P_G, F_G]
    const float* __restrict__ Bias,  // [P_G]
    float* __restrict__ Out)         // [B_G, NX_G, NY_G]
{
    const int lane = threadIdx.x & 31;
    const int wave = threadIdx.x >> 5;
    const int row  = lane & 15;   // M index in A frag; N index in B/C frags
    const int kh   = lane >> 4;   // which K half-pair this lane holds

    const int m0     = (blockIdx.y * 8 + wave) * 16;  // batch tile origin
    const int n0base = blockIdx.x * 64;               // pixel strip origin

    // ---- Load all A fragments once (reused for 4 N-subtiles) ----
    const float* Arow = A + (size_t)(m0 + row) * F_G + 2 * kh;
    v2f afrag[16];
#pragma unroll
    for (int ks = 0; ks < 16; ++ks)
        afrag[ks] = *(const v2f*)(Arow + 4 * ks);

    v8f acc[4] = {};

    // ---- 4 N-subtiles of 16 pixels; 16 K-steps each ----
#pragma unroll
    for (int t = 0; t < 4; ++t) {
        int p = n0base + t * 16 + row;
        if (p > P_G - 1) p = P_G - 1;   // clamp loads: EXEC must stay all-1s for WMMA
        const float* Wrow = W + (size_t)p * F_G + 2 * kh;
#pragma unroll
        for (int ks = 0; ks < 16; ++ks) {
            v2f bfrag = *(const v2f*)(Wrow + 4 * ks);
            // 8 args: (neg_a, A, neg_b, B, c_mod, C, reuse_a, reuse_b)
            acc[t] = __builtin_amdgcn_wmma_f32_16x16x4_f32(
                false, afrag[ks], false, bfrag, (short)0, acc[t], false, false);
        }
    }

    // ---- Epilogue: bias add + transposed scatter store ----
    // C frag layout: VGPR r of lane L holds (M = r + 8*(L/16), N = L%16).
#pragma unroll
    for (int t = 0; t < 4; ++t) {
        const int n = n0base + t * 16 + row;   // this lane's pixel index p
        if (n < P_G) {
            const float bias = Bias[n];
            const int x = n % NX_G;
            const int y = n / NX_G;
            float* outp = Out + (size_t)x * NY_G + y;
#pragma unroll
            for (int r = 0; r < 8; ++r) {
                const int b = m0 + r + 8 * kh;
                outp[(size_t)b * P_G] = acc[t][r] + bias;
            }
        }
    }
}

extern "C" void kernel_launch(void* const* d_in, const int* in_sizes, int n_in,
                              void* d_out, int out_size, void* d_ws, size_t ws_size,
                              hipStream_t stream) {
    (void)in_sizes; (void)n_in; (void)out_size; (void)d_ws; (void)ws_size;
    const float* A    = (const float*)d_in[0];  // inputs [8192, 64]
    const float* W    = (const float*)d_in[1];  // W      [5494, 64]
    const float* Bias = (const float*)d_in[2];  // b      [5494]
    float* Out        = (float*)d_out;          // [8192, 82, 67, 1]

    dim3 block(256);                            // 8 waves (wave32)
    dim3 grid((P_G + 63) / 64,                  // 86 pixel strips
              B_G / (16 * 8));                  // 64 batch blocks (128 rows each)
    lasso_pix_gemm<<<grid, block, 0, stream>>>(A, W, Bias, Out);
}